// Module_GAT_AOUP_split_49254684950643
// MI455X (gfx1250) — compile-verified
//
#include <hip/hip_runtime.h>
#include <math.h>

// ---------------------------------------------------------------------------
// Problem constants (reference: B=4, N=512, D_IN=4, H=128, D_V=32, D_ATT=4)
// ---------------------------------------------------------------------------
#define BB     4
#define NN     512
#define BN     (BB*NN)      // 2048
#define HH     128
#define DV     32
#define NHEAD  4
#define MASKC  1.0e4f

typedef __attribute__((ext_vector_type(2))) float v2f;
typedef __attribute__((ext_vector_type(8))) float v8f;
typedef __attribute__((ext_vector_type(4))) unsigned int v4u;
typedef __attribute__((ext_vector_type(4))) int v4i;
typedef __attribute__((ext_vector_type(8))) int v8i;

__device__ __forceinline__ float sigm(float x) {
    return 1.0f / (1.0f + __expf(-x));
}

#if defined(__gfx1250__) && __has_builtin(__builtin_amdgcn_tensor_load_to_lds)
#define ATT_USE_TDM 1
#else
#define ATT_USE_TDM 0
#endif

#if ATT_USE_TDM
// ---------------------------------------------------------------------------
// Issue one Tensor-Data-Mover load: global 2D tile (data_size = 4B) -> LDS.
// D# built per CDNA5 ISA.  1-D tiles pass
// tile_d1 = 0.  LDS padding (pad every 2^? dwords) comes from the D# fields.
// Descriptor values are wave-uniform -> land in SGPRs.
// ---------------------------------------------------------------------------
__device__ __forceinline__ void tdm_load_to_lds(
    unsigned lds_byte_off, const void* gptr,
    unsigned tile_d0, unsigned tile_d1,
    unsigned tensor_d0, unsigned tensor_d1,
    unsigned long long stride0_elems,
    int pad_en, unsigned pad_interval_code, unsigned pad_amount_code)
{
    unsigned long long ga = (unsigned long long)(size_t)gptr;
    v4u g0;
    g0[0] = 1u;                                        // count=1 (valid), user mode
    g0[1] = lds_byte_off;                              // lds_addr (bytes)
    g0[2] = (unsigned)(ga & 0xFFFFFFFFu);              // global_addr[31:0]
    g0[3] = (unsigned)((ga >> 32) & 0x01FFFFFFu)       // global_addr[56:32]
          | (2u << 30);                                // type = 2 ("image")
    v8i g1;
    unsigned dw0 = (2u << 16)                          // data_size = 4 bytes
                 | (pad_en ? (1u << 20) : 0u)          // pad_enable
                 | ((pad_interval_code & 7u) << 22)
                 | ((pad_amount_code & 127u) << 25);
    g1[0] = (int)dw0;                                  // wg_mask=0 (not in cluster)
    g1[1] = (int)((tensor_d0 & 0xFFFFu) << 16);        // td0[15:0] -> bits[63:48]
    g1[2] = (int)(((tensor_d0 >> 16) & 0xFFFFu)        // td0[31:16]
          | ((tensor_d1 & 0xFFFFu) << 16));            // td1[15:0]
    g1[3] = (int)(((tensor_d1 >> 16) & 0xFFFFu)        // td1[31:16]
          | ((tile_d0 & 0xFFFFu) << 16));              // tile_dim0
    g1[4] = (int)(tile_d1 & 0xFFFFu);                  // tile_dim1 (0 => 1-D); tile_dim2=0
    g1[5] = (int)(unsigned)(stride0_elems & 0xFFFFFFFFu);   // dim0 stride [31:0]
    g1[6] = (int)(unsigned)((stride0_elems >> 32) & 0xFFFFu); // stride [47:32]; dim1 stride=0
    g1[7] = 0;
    v4i z4 = {0, 0, 0, 0};
#if defined(__clang_major__) && (__clang_major__ <= 22)
    // ROCm 7.2 (AMD clang-22): 5-arg form
    __builtin_amdgcn_tensor_load_to_lds(g0, g1, z4, z4, 0);
#else
    // upstream clang-23 (therock headers): 6-arg form
    v8i z8 = {0, 0, 0, 0, 0, 0, 0, 0};
    __builtin_amdgcn_tensor_load_to_lds(g0, g1, z4, z4, z8, 0);
#endif
}
#endif // ATT_USE_TDM

// ---------------------------------------------------------------------------
// Generic f32 WMMA GEMM:  C[M,N] = act( A[M,K] @ W[N,K]^T + bias )  (+= if acc)
// One wave per 16x16 C tile, K stepped by 4 via V_WMMA_F32_16X16X4_F32.
// M,N multiples of 16; K multiple of 4.  A row stride lda, W row stride ldw.
// ---------------------------------------------------------------------------
__global__ void wmma_gemm_kernel(const float* __restrict__ A, int lda,
                                 const float* __restrict__ W, int ldw,
                                 const float* __restrict__ bias,
                                 float* __restrict__ C, int ldc,
                                 int K, int act, int accumulate)
{
    const int tn   = blockIdx.x;        // N tile
    const int tm   = blockIdx.y;        // M tile
    const int lane = threadIdx.x;       // 0..31
    const int half = lane >> 4;         // 0 | 1
    const int l15  = lane & 15;
    const int row  = tm * 16 + l15;     // A row for this lane
    const int col  = tn * 16 + l15;     // B/C column for this lane
    const int kb   = half * 2;          // lane's K sub-offset {0,2}

    v8f acc = {};
    if (accumulate) {
        #pragma unroll
        for (int r = 0; r < 8; ++r) {
            int crow = tm * 16 + half * 8 + r;
            acc[r] = C[crow * ldc + col];
        }
    }

    for (int k = 0; k < K; k += 4) {
        v2f a, b;
        a.x = A[row * lda + k + kb];
        a.y = A[row * lda + k + kb + 1];
        b.x = W[col * ldw + k + kb];
        b.y = W[col * ldw + k + kb + 1];
        acc = __builtin_amdgcn_wmma_f32_16x16x4_f32(
                  false, a, false, b, (short)0, acc, false, false);
    }

    const float bv = bias ? bias[col] : 0.0f;
    #pragma unroll
    for (int r = 0; r < 8; ++r) {
        int crow = tm * 16 + half * 8 + r;
        float v = acc[r] + bv;
        if (act == 1) v = fmaxf(v, 0.0f);
        C[crow * ldc + col] = v;
    }
}

// ---------------------------------------------------------------------------
// LSTM gate nonlinearity: G[2048,512] holds x@Wih^T + h@Whh^T (no bias yet).
// Gate order i,f,g,o (torch).  Writes b (=h2) and c2.
// ---------------------------------------------------------------------------
__global__ void lstm_gates_kernel(const float* __restrict__ G,
                                  const float* __restrict__ b_ih,
                                  const float* __restrict__ b_hh,
                                  const float* __restrict__ cell,
                                  float* __restrict__ bmat,
                                  float* __restrict__ cmat)
{
    int idx = blockIdx.x * blockDim.x + threadIdx.x;
    if (idx >= BN * HH) return;
    int row = idx >> 7, h = idx & 127;
    const float* g = G + row * (4 * HH);
    float gi = g[h]          + b_ih[h]          + b_hh[h];
    float gf = g[HH + h]     + b_ih[HH + h]     + b_hh[HH + h];
    float gg = g[2*HH + h]   + b_ih[2*HH + h]   + b_hh[2*HH + h];
    float go = g[3*HH + h]   + b_ih[3*HH + h]   + b_hh[3*HH + h];
    float c2 = sigm(gf) * cell[idx] + sigm(gi) * tanhf(gg);
    float h2 = sigm(go) * tanhf(c2);
    bmat[idx] = h2;
    cmat[idx] = c2;
}

// ---------------------------------------------------------------------------
// Fused attention-MLP + heads einsum.
//   hidden_ij = relu(K1[i] + Q1[j] + R[b,i,j]*w1r)          (b1 folded in Q1)
//   logit_a   = hidden . W2[a] + b2[a];  w = sigmoid(logit - eye*1e4)/N
//   heads[a,b,i,:] += w * V[b,j,:]       via V_WMMA_F32_16X16X4_F32
//
// One 128-thread workgroup (4 waves) per (batch, 16-row i-tile).
// All of Q1[b] (64KB), V[b] (64KB) and the 16x512 R tile (32KB, bank-padded
// row stride 516 dwords) staged in LDS via the Tensor Data Mover (wave 0,
// 3 descriptors, s_wait_tensorcnt), fallback = manual coalesced staging.
// Inner loop then performs zero global-memory accesses.
// Each wave owns a disjoint 128-wide j range; partials reduced via LDS in
// fixed order -> deterministic.
// ---------------------------------------------------------------------------
#define RS_STRIDE 516                      // 512 data + 2x2 dword pad per row
#define SM_QS   0                          // 512*32 = 16384
#define SM_VS   16384                      // 512*32 = 16384
#define SM_RS   32768                      // 16*516 = 8256
#define SM_W2   (SM_RS + 16*RS_STRIDE)     // 41024, 4*32 = 128
#define SM_W1R  (SM_W2 + 128)              // 41152, 32
#define SM_B2   (SM_W1R + 32)              // 41184, 4 (padded to 16)
#define SM_PART (SM_B2 + 16)               // 41200, 4 waves * 2048 = 8192
#define SM_TOTAL_FLOATS (SM_PART + 8192)   // 49392 floats = 197568 bytes

__device__ __forceinline__ int rs_idx(int i, int j) {
    return i * RS_STRIDE + j + ((j >> 8) << 1);   // matches TDM pad(256,+2)
}

__global__ void att_heads_kernel(const float* __restrict__ K1,    // [BN,32]
                                 const float* __restrict__ Q1,    // [BN,32]
                                 const float* __restrict__ V,     // [BN,32]
                                 const float* __restrict__ R,     // [B,N,N]
                                 const float* __restrict__ attW1, // [32,65]
                                 const float* __restrict__ attW2, // [4,32]
                                 const float* __restrict__ attb2, // [4]
                                 float* __restrict__ heads)       // [4,BN,32]
{
    extern __shared__ float smem[];
    float* Qs   = smem + SM_QS;
    float* Vs   = smem + SM_VS;
    float* Rs   = smem + SM_RS;
    float* W2s  = smem + SM_W2;
    float* w1r  = smem + SM_W1R;
    float* b2s  = smem + SM_B2;
    float* part = smem + SM_PART;

    const int blk   = blockIdx.x;        // 0..127
    const int b     = blk >> 5;          // batch
    const int itile = blk & 31;
    const int i0    = itile * 16;
    const int tid   = threadIdx.x;       // 0..127

    // ---- stage big tiles into LDS ----
#if ATT_USE_TDM
    if (tid < 32) {   // wave 0 issues the DMA descriptors
        // Q1[b] : 1-D, 16384 f32
        tdm_load_to_lds((unsigned)(size_t)Qs, Q1 + (size_t)b * NN * DV,
                        NN * DV, 0, NN * DV, 1, NN * DV, 0, 0, 0);
        // V[b]  : 1-D, 16384 f32
        tdm_load_to_lds((unsigned)(size_t)Vs, V + (size_t)b * NN * DV,
                        NN * DV, 0, NN * DV, 1, NN * DV, 0, 0, 0);
        // R tile: 2-D 16 rows x 512 cols, row stride 512, LDS pad 2 dwords
        // every 256 dwords (pad_interval code 7 = 256, pad_amount code 1 = 2)
        tdm_load_to_lds((unsigned)(size_t)Rs,
                        R + ((size_t)b * NN + i0) * NN,
                        NN, 16, NN, 16, NN, 1, 7, 1);
        __builtin_amdgcn_s_wait_tensorcnt(0);
    }
#else
    for (int t = tid; t < NN * DV; t += 128) {
        Qs[t] = Q1[(size_t)b * NN * DV + t];
        Vs[t] = V[(size_t)b * NN * DV + t];
    }
    for (int t = tid; t < 16 * NN; t += 128) {
        int i = t >> 9, j = t & (NN - 1);
        Rs[rs_idx(i, j)] = R[((size_t)b * NN + i0 + i) * NN + j];
    }
#endif
    // small operands (all threads, cheap)
    if (tid < NHEAD * DV) W2s[tid] = attW2[tid];
    if (tid < DV)         w1r[tid] = attW1[tid * 65 + 64];   // R column of W1
    if (tid < NHEAD)      b2s[tid] = attb2[tid];
    __syncthreads();

    const int wave = tid >> 5;
    const int lane = tid & 31;
    const int half = lane >> 4;
    const int i    = lane & 15;
    const int koff = half * 2;
    const int gi   = i0 + i;                       // global row index
    const int jwbase = wave * (NN / 4);            // 128 j's per wave

    // K1 row for this lane is fixed -> hoist into registers (no LDS conflicts)
    float kq[DV];
    #pragma unroll
    for (int e = 0; e < DV; ++e)
        kq[e] = K1[((size_t)b * NN + gi) * DV + e];

    v8f acc[NHEAD][2];
    #pragma unroll
    for (int a = 0; a < NHEAD; ++a) { acc[a][0] = (v8f){}; acc[a][1] = (v8f){}; }

    for (int j0 = 0; j0 < NN / 4; j0 += 4) {
        const int jl = jwbase + j0 + koff;         // lane's first j (of 2)
        float wv[NHEAD][2];
        #pragma unroll
        for (int u = 0; u < 2; ++u) {
            const int j = jl + u;
            const float r = Rs[rs_idx(i, j)];
            const float* qp = Qs + j * DV;
            float l0 = 0.f, l1 = 0.f, l2 = 0.f, l3 = 0.f;
            #pragma unroll
            for (int e = 0; e < DV; ++e) {
                float t = fmaf(r, w1r[e], kq[e] + qp[e]);
                t = fmaxf(t, 0.0f);
                l0 = fmaf(t, W2s[e],          l0);
                l1 = fmaf(t, W2s[DV + e],     l1);
                l2 = fmaf(t, W2s[2 * DV + e], l2);
                l3 = fmaf(t, W2s[3 * DV + e], l3);
            }
            const float mask = (gi == j) ? MASKC : 0.0f;
            const float inv_n = 1.0f / (float)NN;
            wv[0][u] = sigm(l0 + b2s[0] - mask) * inv_n;
            wv[1][u] = sigm(l1 + b2s[1] - mask) * inv_n;
            wv[2][u] = sigm(l2 + b2s[2] - mask) * inv_n;
            wv[3][u] = sigm(l3 + b2s[3] - mask) * inv_n;
        }
        // B tiles from V (shared across heads): col = lane&15 (+16 for tile 1)
        v2f bt0, bt1;
        bt0.x = Vs[jl * DV + i];            bt0.y = Vs[(jl + 1) * DV + i];
        bt1.x = Vs[jl * DV + 16 + i];       bt1.y = Vs[(jl + 1) * DV + 16 + i];
        #pragma unroll
        for (int a = 0; a < NHEAD; ++a) {
            v2f at; at.x = wv[a][0]; at.y = wv[a][1];
            acc[a][0] = __builtin_amdgcn_wmma_f32_16x16x4_f32(
                            false, at, false, bt0, (short)0, acc[a][0], false, false);
            acc[a][1] = __builtin_amdgcn_wmma_f32_16x16x4_f32(
                            false, at, false, bt1, (short)0, acc[a][1], false, false);
        }
    }

    // ---- write wave partials (C/D layout -> [row][d]) ----
    #pragma unroll
    for (int a = 0; a < NHEAD; ++a)
        #pragma unroll
        for (int nt = 0; nt < 2; ++nt)
            #pragma unroll
            for (int r = 0; r < 8; ++r) {
                int prow = half * 8 + r;
                int d = nt * 16 + i;
                part[wave * 2048 + a * 512 + prow * DV + d] = acc[a][nt][r];
            }
    __syncthreads();

    // ---- deterministic 4-way reduction, write heads ----
    for (int t = tid; t < NHEAD * 16 * DV; t += 128) {
        float s = part[t] + part[2048 + t] + part[4096 + t] + part[6144 + t];
        int a = t >> 9, rem = t & 511, r = rem >> 5, d = rem & 31;
        heads[(size_t)a * BN * DV + (b * NN + i0 + r) * DV + d] = s;
    }
}

// ---------------------------------------------------------------------------
// ci[a][row][0:64] = [ V[row][0:32] | heads[a][row][0:32] ]
// ---------------------------------------------------------------------------
__global__ void pack_ci_kernel(const float* __restrict__ V,
                               const float* __restrict__ heads,
                               float* __restrict__ ci)
{
    int idx = blockIdx.x * blockDim.x + threadIdx.x;
    if (idx >= NHEAD * BN * 64) return;
    int a = idx / (BN * 64);
    int rem = idx - a * (BN * 64);
    int row = rem >> 6, dd = rem & 63;
    ci[idx] = (dd < DV) ? V[row * DV + dd]
                        : heads[(size_t)a * BN * DV + row * DV + (dd - DV)];
}

// ---------------------------------------------------------------------------
// mu = d . Wm + bm ; sig = softplus(d . Ws + bs)  for one head
// ---------------------------------------------------------------------------
__global__ void mu_sig_kernel(const float* __restrict__ D,      // [BN,64]
                              const float* __restrict__ muW, const float* __restrict__ mub,
                              const float* __restrict__ sgW, const float* __restrict__ sgb,
                              float* __restrict__ outmu, float* __restrict__ outsig)
{
    int row = blockIdx.x * blockDim.x + threadIdx.x;
    if (row >= BN) return;
    const float* d = D + row * 64;
    float m = mub[0], s = sgb[0];
    #pragma unroll 8
    for (int e = 0; e < 64; ++e) {
        m = fmaf(d[e], muW[e], m);
        s = fmaf(d[e], sgW[e], s);
    }
    outmu[row]  = m;
    outsig[row] = (s > 20.0f) ? s : log1pf(__expf(s));
}

// ---------------------------------------------------------------------------
// Host-side orchestration
// ---------------------------------------------------------------------------
extern "C" void kernel_launch(void* const* d_in, const int* in_sizes, int n_in,
                              void* d_out, int out_size, void* d_ws, size_t ws_size,
                              hipStream_t stream) {
    (void)in_sizes; (void)n_in; (void)out_size; (void)ws_size;

    const float* x      = (const float*)d_in[0];   // [BN,4]
    const float* hidden = (const float*)d_in[1];   // [BN,128]
    const float* cell   = (const float*)d_in[2];   // [BN,128]
    const float* R      = (const float*)d_in[3];   // [B,N,N]
    const float* W_ih   = (const float*)d_in[4];   // [512,4]
    const float* W_hh   = (const float*)d_in[5];   // [512,128]
    const float* b_ih   = (const float*)d_in[6];
    const float* b_hh   = (const float*)d_in[7];
    const float* kW1 = (const float*)d_in[8],  *kb1 = (const float*)d_in[9];
    const float* kW2 = (const float*)d_in[10], *kb2 = (const float*)d_in[11];
    const float* qW1 = (const float*)d_in[12], *qb1 = (const float*)d_in[13];
    const float* qW2 = (const float*)d_in[14], *qb2 = (const float*)d_in[15];
    const float* vW1 = (const float*)d_in[16], *vb1 = (const float*)d_in[17];
    const float* vW2 = (const float*)d_in[18], *vb2 = (const float*)d_in[19];
    const float* attW1 = (const float*)d_in[20];   // [32,65]
    const float* attb1 = (const float*)d_in[21];
    const float* attW2 = (const float*)d_in[22];   // [4,32]
    const float* attb2 = (const float*)d_in[23];

    // Workspace layout (floats)
    float* ws    = (float*)d_ws;
    float* G     = ws;                    // BN*512      = 1048576
    float* kmat  = G    + BN * 512;       // BN*32
    float* qmat  = kmat + BN * DV;
    float* vmat  = qmat + BN * DV;
    float* hid   = vmat + BN * DV;        // BN*64
    float* K1    = hid  + BN * 64;        // BN*32
    float* Q1    = K1   + BN * DV;        // BN*32
    float* heads = Q1   + BN * DV;        // 4*BN*32
    float* ci    = heads + NHEAD * BN * DV;   // 4*BN*64
    float* dmat  = ci    + NHEAD * BN * 64;   // 4*BN*64

    // Output layout: mu0,sig0,...,mu3,sig3 ([B,N] each), then b, then c
    float* out  = (float*)d_out;
    float* bmat = out + 8 * BN;               // [BN,128]
    float* cmat = bmat + BN * HH;             // [BN,128]

    // 1) LSTM gates pre-activation: G = h@Whh^T ; G += x@Wih^T
    wmma_gemm_kernel<<<dim3(512/16, BN/16), 32, 0, stream>>>(
        hidden, HH, W_hh, HH, nullptr, G, 512, HH, 0, 0);
    wmma_gemm_kernel<<<dim3(512/16, BN/16), 32, 0, stream>>>(
        x, 4, W_ih, 4, nullptr, G, 512, 4, 0, 1);

    // 2) Gate nonlinearity -> b (h2) and c2 directly into d_out
    lstm_gates_kernel<<<(BN*HH + 255)/256, 256, 0, stream>>>(
        G, b_ih, b_hh, cell, bmat, cmat);

    // 3) k/q/v two-layer MLPs (relu hidden), reusing `hid`
    wmma_gemm_kernel<<<dim3(64/16, BN/16), 32, 0, stream>>>(bmat, HH, kW1, HH, kb1, hid, 64, HH, 1, 0);
    wmma_gemm_kernel<<<dim3(DV/16, BN/16), 32, 0, stream>>>(hid, 64, kW2, 64, kb2, kmat, DV, 64, 0, 0);
    wmma_gemm_kernel<<<dim3(64/16, BN/16), 32, 0, stream>>>(bmat, HH, qW1, HH, qb1, hid, 64, HH, 1, 0);
    wmma_gemm_kernel<<<dim3(DV/16, BN/16), 32, 0, stream>>>(hid, 64, qW2, 64, qb2, qmat, DV, 64, 0, 0);
    wmma_gemm_kernel<<<dim3(64/16, BN/16), 32, 0, stream>>>(bmat, HH, vW1, HH, vb1, hid, 64, HH, 1, 0);
    wmma_gemm_kernel<<<dim3(DV/16, BN/16), 32, 0, stream>>>(hid, 64, vW2, 64, vb2, vmat, DV, 64, 0, 0);

    // 4) Factor attention first layer through the concat:
    //    K1 = k@W1[:, :32]^T ; Q1 = q@W1[:, 32:64]^T + b1
    wmma_gemm_kernel<<<dim3(DV/16, BN/16), 32, 0, stream>>>(kmat, DV, attW1,      65, nullptr, K1, DV, DV, 0, 0);
    wmma_gemm_kernel<<<dim3(DV/16, BN/16), 32, 0, stream>>>(qmat, DV, attW1 + 32, 65, attb1,  Q1, DV, DV, 0, 0);

    // 5) Fused attention weights + heads einsum (TDM staging + WMMA accum)
    att_heads_kernel<<<BB * (NN/16), 128, SM_TOTAL_FLOATS * sizeof(float), stream>>>(
        K1, Q1, vmat, R, attW1, attW2, attb2, heads);

    // 6) Decoder per head: ci = [v|heads_a] ; d = ci@Wd^T + bd ; mu/sig
    pack_ci_kernel<<<(NHEAD*BN*64 + 255)/256, 256, 0, stream>>>(vmat, heads, ci);
    for (int a = 0; a < NHEAD; ++a) {
        const float* decW = (const float*)d_in[24 + 2*a];
        const float* decb = (const float*)d_in[25 + 2*a];
        wmma_gemm_kernel<<<dim3(64/16, BN/16), 32, 0, stream>>>(
            ci + (size_t)a * BN * 64, 64, decW, 64, decb,
            dmat + (size_t)a * BN * 64, 64, 64, 0, 0);
    }
    for (int a = 0; a < NHEAD; ++a) {
        const float* muW = (const float*)d_in[32 + 2*a];
        const float* mub = (const float*)d_in[33 + 2*a];
        const float* sgW = (const float*)d_in[40 + 2*a];
        const float* sgb = (const float*)d_in[41 + 2*a];
        mu_sig_kernel<<<(BN + 255)/256, 256, 0, stream>>>(
            dmat + (size_t)a * BN * 64, muW, mub, sgW, sgb,
            out + a * 2 * BN, out + a * 2 * BN + BN);
    }
}